// StudentNet_54400055771739
// MI455X (gfx1250) — compile-verified
//
#include <hip/hip_runtime.h>
#include <math.h>

// ---------- constants ----------
#define Bb 16
#define Nn 64
#define Pp 6
#define Dd 8
#define Ee 128
#define Hh 8
#define Hd 16
#define Ii 64
#define Oo 128
#define NPROTO 256      // D * N_PROTO = 8*32
#define S  512          // N*D
#define ROWS 8192       // B*S
#define KV 1024         // 2*S

typedef float v2f __attribute__((ext_vector_type(2)));
typedef float v8f __attribute__((ext_vector_type(8)));
typedef int   int4v __attribute__((ext_vector_type(4)));
typedef __attribute__((address_space(1))) int4v* as1_int4_ptr;
typedef __attribute__((address_space(3))) int4v* as3_int4_ptr;

#if defined(__has_builtin)
#if __has_builtin(__builtin_amdgcn_global_load_async_to_lds_b128) && \
    __has_builtin(__builtin_amdgcn_s_wait_asynccnt)
#define HAVE_ASYNC_LDS 1
#endif
#endif

__device__ __forceinline__ v8f wmma4(v2f a, v2f b, v8f c) {
    // D = A(16x4 f32) * B(4x16 f32) + C(16x16 f32)
    return __builtin_amdgcn_wmma_f32_16x16x4_f32(false, a, false, b, (short)0, c,
                                                 false, false);
}

__device__ __forceinline__ v8f v8zero() {
    v8f z = {0.f, 0.f, 0.f, 0.f, 0.f, 0.f, 0.f, 0.f};
    return z;
}

// ---------------------------------------------------------------------------
// K1: instance-norm rows, cosine argmax vs prototypes, both embeddings.
// One thread per row of x_flat (8192 rows, K=6).  Prototypes + weights in LDS.
// ---------------------------------------------------------------------------
__global__ void embed_kernel(const float* __restrict__ x,
                             const float* __restrict__ protos,
                             const float* __restrict__ W_emb,
                             const float* __restrict__ b_emb,
                             const float* __restrict__ W_proto,
                             const float* __restrict__ b_proto,
                             float* __restrict__ ie, float* __restrict__ pe) {
    __shared__ float pn [NPROTO * Pp];
    __shared__ float pl2[NPROTO * Pp];
    __shared__ float sWe[Ee * Pp];
    __shared__ float sWp[Ee * Pp];
    __shared__ float sbe[Ee];
    __shared__ float sbp[Ee];
    int t = threadIdx.x;           // 256 threads

    // normalize prototype row t (inorm + l2)
    {
        float v[Pp]; float mn = 0.f;
        #pragma unroll
        for (int p = 0; p < Pp; ++p) { v[p] = protos[t * Pp + p]; mn += v[p]; }
        mn *= (1.f / Pp);
        float var = 0.f;
        #pragma unroll
        for (int p = 0; p < Pp; ++p) { float d = v[p] - mn; var += d * d; }
        var *= (1.f / Pp);
        float inv = rsqrtf(var + 1e-5f);
        float n2 = 0.f;
        #pragma unroll
        for (int p = 0; p < Pp; ++p) {
            v[p] = (v[p] - mn) * inv;
            pn[t * Pp + p] = v[p];
            n2 += v[p] * v[p];
        }
        float nrm = fmaxf(sqrtf(n2), 1e-12f);
        #pragma unroll
        for (int p = 0; p < Pp; ++p) pl2[t * Pp + p] = v[p] / nrm;
    }
    for (int i = t; i < Ee * Pp; i += 256) { sWe[i] = W_emb[i]; sWp[i] = W_proto[i]; }
    if (t < Ee) { sbe[t] = b_emb[t]; sbp[t] = b_proto[t]; }
    __syncthreads();

    int row = blockIdx.x * 256 + t;            // 0..8191
    int d  = row & (Dd - 1);
    int bn = row >> 3;                          // b*N + n

    float xv[Pp]; float mn = 0.f;
    #pragma unroll
    for (int p = 0; p < Pp; ++p) { xv[p] = x[(bn * Pp + p) * Dd + d]; mn += xv[p]; }
    mn *= (1.f / Pp);
    float var = 0.f;
    #pragma unroll
    for (int p = 0; p < Pp; ++p) { float dd = xv[p] - mn; var += dd * dd; }
    var *= (1.f / Pp);
    float inv = rsqrtf(var + 1e-5f);
    float xn[Pp], xl[Pp]; float n2 = 0.f;
    #pragma unroll
    for (int p = 0; p < Pp; ++p) { xn[p] = (xv[p] - mn) * inv; n2 += xn[p] * xn[p]; }
    float nrm = fmaxf(sqrtf(n2), 1e-12f);
    #pragma unroll
    for (int p = 0; p < Pp; ++p) xl[p] = xn[p] / nrm;

    int best = 0; float bs = -3.0e38f;
    for (int j = 0; j < NPROTO; ++j) {
        float s = 0.f;
        #pragma unroll
        for (int p = 0; p < Pp; ++p) s += xl[p] * pl2[j * Pp + p];
        if (s > bs) { bs = s; best = j; }
    }
    float sel[Pp];
    #pragma unroll
    for (int p = 0; p < Pp; ++p) sel[p] = pn[best * Pp + p];

    for (int e = 0; e < Ee; ++e) {
        float s1 = sbe[e], s2 = sbp[e];
        #pragma unroll
        for (int p = 0; p < Pp; ++p) {
            s1 += xn[p]  * sWe[e * Pp + p];
            s2 += sel[p] * sWp[e * Pp + p];
        }
        ie[row * Ee + e] = s1;
        pe[row * Ee + e] = s2;
    }
}

// ---------------------------------------------------------------------------
// K2: fused QKV projection.  Virtual A = per-batch [input_embed; proto_embed]
// (16384 x 128).  grid.y in 0..23 selects 16 rows of in_proj_w (Wq/Wk/Wv).
// One wave per 16x16 tile; f32 WMMA, K=128 -> 32 v_wmma per tile.
// ---------------------------------------------------------------------------
__global__ void qkv_kernel(const float* __restrict__ ie, const float* __restrict__ pe,
                           const float* __restrict__ W, const float* __restrict__ bias,
                           float* __restrict__ qb, float* __restrict__ kb,
                           float* __restrict__ vb) {
    int lane = threadIdx.x;
    int mt = blockIdx.x;                 // 0..1023
    int nt = blockIdx.y;                 // 0..23
    if (nt < 8 && (mt & 63) >= 32) return;   // q only comes from the input half
    int m = lane & 15, hi = lane >> 4, koff = hi * 2;

    int r = mt * 16 + m;
    int b = r >> 10, s = r & (KV - 1);
    const float* arow = (s < S) ? (ie + (b * S + s) * Ee)
                                : (pe + (b * S + (s - S)) * Ee);
    const float* wrow = W + (nt * 16 + m) * Ee;

    v8f c = v8zero();
    #pragma unroll 8
    for (int k0 = 0; k0 < Ee; k0 += 4) {
        v2f a = *(const v2f*)(arow + k0 + koff);
        v2f w = *(const v2f*)(wrow + k0 + koff);
        c = wmma4(a, w, c);
    }
    float bv = bias[nt * 16 + m];        // this lane's column n == m
    #pragma unroll
    for (int i = 0; i < 8; ++i) {
        float val = c[i] + bv;
        int rg = mt * 16 + i + 8 * hi;
        int bb = rg >> 10, ss = rg & (KV - 1);
        if (nt < 8) {
            qb[((bb * Hh + nt) * S + ss) * Hd + m] = val;
        } else if (nt < 16) {
            kb[((bb * Hh + (nt - 8)) * KV + ss) * Hd + m] = val;
        } else {
            vb[((bb * Hh + (nt - 16)) * KV + ss) * Hd + m] = val;
        }
    }
}

// ---------------------------------------------------------------------------
// K3: flash attention.  One wave per (b,h,q-tile of 16).  Per 16-key tile:
// 4 wmma for S = q k^T, online softmax (shfl within 16-lane halves -- matches
// the wave32 C-tile layout), LDS transpose of P, 4 wmma for P V.
// ---------------------------------------------------------------------------
__global__ void attn_kernel(const float* __restrict__ qbuf,
                            const float* __restrict__ kbuf,
                            const float* __restrict__ vbuf,
                            float* __restrict__ o) {
    __shared__ float plds[16 * 17];
    int lane = threadIdx.x;
    int qt = blockIdx.x;                 // 0..31
    int bh = blockIdx.y;                 // 0..127
    int m = lane & 15, hi = lane >> 4, koff = hi * 2;

    const float* qp = qbuf + (size_t)bh * S * Hd;
    const float* kp = kbuf + (size_t)bh * KV * Hd;
    const float* vp = vbuf + (size_t)bh * KV * Hd;

    const float* qrow = qp + (qt * 16 + m) * Hd;
    v2f qa[4];
    #pragma unroll
    for (int i = 0; i < 4; ++i) qa[i] = *(const v2f*)(qrow + i * 4 + koff);

    float mrow[8], lrow[8];
    v8f acc = v8zero();
    #pragma unroll
    for (int i = 0; i < 8; ++i) { mrow[i] = -3.0e38f; lrow[i] = 0.f; }

    for (int kt = 0; kt < KV / 16; ++kt) {
        const float* krow = kp + (kt * 16 + m) * Hd;   // lane's column n == m
        v8f sc8 = v8zero();
        #pragma unroll
        for (int i = 0; i < 4; ++i) {
            v2f kf = *(const v2f*)(krow + i * 4 + koff);
            sc8 = wmma4(qa[i], kf, sc8);
        }
        float sc[8], tm[8];
        #pragma unroll
        for (int i = 0; i < 8; ++i) { sc[i] = sc8[i] * 0.25f; tm[i] = sc[i]; }
        #pragma unroll
        for (int off = 1; off < 16; off <<= 1)
            #pragma unroll
            for (int i = 0; i < 8; ++i)
                tm[i] = fmaxf(tm[i], __shfl_xor(tm[i], off));
        float scale[8];
        #pragma unroll
        for (int i = 0; i < 8; ++i) {
            float mnew = fmaxf(mrow[i], tm[i]);
            scale[i] = __expf(mrow[i] - mnew);
            mrow[i] = mnew;
            sc[i] = __expf(sc[i] - mnew);               // p value
        }
        float rs[8];
        #pragma unroll
        for (int i = 0; i < 8; ++i) rs[i] = sc[i];
        #pragma unroll
        for (int off = 1; off < 16; off <<= 1)
            #pragma unroll
            for (int i = 0; i < 8; ++i)
                rs[i] += __shfl_xor(rs[i], off);
        #pragma unroll
        for (int i = 0; i < 8; ++i) {
            lrow[i] = lrow[i] * scale[i] + rs[i];
            acc[i] *= scale[i];
        }
        __syncthreads();
        #pragma unroll
        for (int i = 0; i < 8; ++i) plds[(i + 8 * hi) * 17 + m] = sc[i];
        __syncthreads();
        const float* vt = vp + kt * 16 * Hd;
        #pragma unroll
        for (int i = 0; i < 4; ++i) {
            int kk = i * 4 + koff;
            v2f pa; pa.x = plds[m * 17 + kk];     pa.y = plds[m * 17 + kk + 1];
            v2f vf; vf.x = vt[kk * Hd + m];       vf.y = vt[(kk + 1) * Hd + m];
            acc = wmma4(pa, vf, acc);
        }
    }
    int b = bh >> 3, h = bh & 7;
    #pragma unroll
    for (int i = 0; i < 8; ++i) {
        int srow = qt * 16 + i + 8 * hi;
        o[(b * S + srow) * Ee + h * Hd + m] = acc[i] / lrow[i];
    }
}

// ---------------------------------------------------------------------------
// Generic WMMA GEMM: C(M,N) = act(A(M,K) @ W(N,K)^T + bias).  1 wave / tile.
// ---------------------------------------------------------------------------
__global__ void gemm_kernel(const float* __restrict__ A, const float* __restrict__ W,
                            const float* __restrict__ bias, float* __restrict__ C,
                            int K, int N, int relu) {
    int lane = threadIdx.x;
    int mt = blockIdx.x, nt = blockIdx.y;
    int m = lane & 15, hi = lane >> 4, koff = hi * 2;
    const float* arow = A + (size_t)(mt * 16 + m) * K;
    const float* wrow = W + (size_t)(nt * 16 + m) * K;
    v8f c = v8zero();
    for (int k0 = 0; k0 < K; k0 += 4) {
        v2f a = *(const v2f*)(arow + k0 + koff);
        v2f w = *(const v2f*)(wrow + k0 + koff);
        c = wmma4(a, w, c);
    }
    float bv = bias[nt * 16 + m];
    #pragma unroll
    for (int i = 0; i < 8; ++i) {
        float val = c[i] + bv;
        if (relu) val = fmaxf(val, 0.f);
        C[(size_t)(mt * 16 + i + 8 * hi) * N + nt * 16 + m] = val;
    }
}

// ---------------------------------------------------------------------------
// LayerNorm(a + b) over rows of 128; one wave per row (4 cols / lane).
// ---------------------------------------------------------------------------
__global__ void lnadd_kernel(const float* __restrict__ a, const float* __restrict__ b,
                             const float* __restrict__ g, const float* __restrict__ be,
                             float* __restrict__ out) {
    int wave = threadIdx.x >> 5, lane = threadIdx.x & 31;
    int row = blockIdx.x * 8 + wave;
    const float* pa = a + (size_t)row * Ee;
    const float* pb = b + (size_t)row * Ee;
    float v[4]; float sum = 0.f;
    #pragma unroll
    for (int i = 0; i < 4; ++i) { v[i] = pa[lane + i * 32] + pb[lane + i * 32]; sum += v[i]; }
    #pragma unroll
    for (int off = 1; off < 32; off <<= 1) sum += __shfl_xor(sum, off);
    float mean = sum * (1.f / Ee);
    float var = 0.f;
    #pragma unroll
    for (int i = 0; i < 4; ++i) { float d = v[i] - mean; var += d * d; }
    #pragma unroll
    for (int off = 1; off < 32; off <<= 1) var += __shfl_xor(var, off);
    var *= (1.f / Ee);
    float inv = rsqrtf(var + 1e-5f);
    #pragma unroll
    for (int i = 0; i < 4; ++i) {
        int cidx = lane + i * 32;
        out[(size_t)row * Ee + cidx] = (v[i] - mean) * inv * g[cidx] + be[cidx];
    }
}

// ---------------------------------------------------------------------------
// Final readout: out(16,128) = Z(16,65536) @ W_out(128,65536)^T + b_out.
// Split-K WMMA (8 n-tiles x 64 k-splits) so W_out streams through HBM once.
// Async path: double-buffered GLOBAL_LOAD_ASYNC_TO_LDS_B128 staging of 64-float
// K-pieces (padded LDS rows, stride 68), pipelined with s_wait_asynccnt.
// ---------------------------------------------------------------------------
#define FK 65536
#define LSTR 68                 // LDS row stride in floats (16B padding)

#ifdef HAVE_ASYNC_LDS
__device__ __forceinline__ void stage_piece(const float* gA, const float* gW,
                                            float* lA, float* lW, int lane) {
    #pragma unroll
    for (int it = 0; it < 8; ++it) {
        int seg = it * 32 + lane;            // 0..255
        int row = seg >> 4;
        int c4  = (seg & 15) * 4;            // float index of 16B segment
        __builtin_amdgcn_global_load_async_to_lds_b128(
            (as1_int4_ptr)(gA + (size_t)row * FK + c4),
            (as3_int4_ptr)(lA + row * LSTR + c4), 0, 0);
        __builtin_amdgcn_global_load_async_to_lds_b128(
            (as1_int4_ptr)(gW + (size_t)row * FK + c4),
            (as3_int4_ptr)(lW + row * LSTR + c4), 0, 0);
    }
}
#endif

__global__ void final_partial(const float* __restrict__ Z, const float* __restrict__ Wout,
                              float* __restrict__ partial) {
    int lane = threadIdx.x;
    int ks = blockIdx.x;                  // 0..63
    int nt = blockIdx.y;                  // 0..7
    int m = lane & 15, koff = (lane >> 4) * 2;
    const float* arow = Z    + (size_t)m * FK + ks * 1024;
    const float* wrow = Wout + (size_t)(nt * 16 + m) * FK + ks * 1024;
    v8f c = v8zero();

#ifdef HAVE_ASYNC_LDS
    __shared__ float lA[2][16 * LSTR];
    __shared__ float lW[2][16 * LSTR];
    const float* gA = Z    + (size_t)0 * FK + ks * 1024;           // rows 0..15
    const float* gW = Wout + (size_t)(nt * 16) * FK + ks * 1024;   // rows nt*16..
    stage_piece(gA, gW, lA[0], lW[0], lane);                       // piece 0 (16 ops)
    for (int p = 0; p < 16; ++p) {
        if (p < 15)
            stage_piece(gA + (p + 1) * 64, gW + (p + 1) * 64,
                        lA[(p + 1) & 1], lW[(p + 1) & 1], lane);
        // async loads complete in order: allow only the newest 16 outstanding
        if (p < 15) __builtin_amdgcn_s_wait_asynccnt(16);
        else        __builtin_amdgcn_s_wait_asynccnt(0);
        asm volatile("" ::: "memory");   // keep LDS reads below the wait
        const float* pA = lA[p & 1];
        const float* pW = lW[p & 1];
        #pragma unroll
        for (int k0 = 0; k0 < 64; k0 += 4) {
            v2f a = *(const v2f*)(pA + m * LSTR + k0 + koff);
            v2f w = *(const v2f*)(pW + m * LSTR + k0 + koff);
            c = wmma4(a, w, c);
        }
    }
#else
    for (int k0 = 0; k0 < 1024; k0 += 4) {
        if ((k0 & 255) == 0) __builtin_prefetch(wrow + k0 + 256, 0, 1);
        v2f a = *(const v2f*)(arow + k0 + koff);
        v2f w = *(const v2f*)(wrow + k0 + koff);
        c = wmma4(a, w, c);
    }
#endif

    float* pt = partial + (size_t)(nt * 64 + ks) * 256;
    #pragma unroll
    for (int i = 0; i < 8; ++i) pt[i * 32 + lane] = c[i];
}

__global__ void final_reduce(const float* __restrict__ partial,
                             const float* __restrict__ b_out, float* __restrict__ out) {
    int t = threadIdx.x;                  // 256
    int nt = blockIdx.x;                  // 0..7
    float s = 0.f;
    for (int ks = 0; ks < 64; ++ks) s += partial[(size_t)(nt * 64 + ks) * 256 + t];
    int i = t >> 5, lane = t & 31;
    int m = i + 8 * (lane >> 4), n = lane & 15;
    out[m * Oo + nt * 16 + n] = s + b_out[nt * 16 + n];
}

// ---------------------------------------------------------------------------
extern "C" void kernel_launch(void* const* d_in, const int* in_sizes, int n_in,
                              void* d_out, int out_size, void* d_ws, size_t ws_size,
                              hipStream_t stream) {
    const float* x        = (const float*)d_in[0];
    const float* protos   = (const float*)d_in[1];
    const float* W_emb    = (const float*)d_in[2];
    const float* b_emb    = (const float*)d_in[3];
    const float* W_proto  = (const float*)d_in[4];
    const float* b_proto  = (const float*)d_in[5];
    const float* in_w     = (const float*)d_in[6];
    const float* in_b     = (const float*)d_in[7];
    const float* op_w     = (const float*)d_in[8];
    const float* op_b     = (const float*)d_in[9];
    const float* W1       = (const float*)d_in[10];
    const float* b1       = (const float*)d_in[11];
    const float* W2       = (const float*)d_in[12];
    const float* b2       = (const float*)d_in[13];
    const float* g1       = (const float*)d_in[14];
    const float* be1      = (const float*)d_in[15];
    const float* g2       = (const float*)d_in[16];
    const float* be2      = (const float*)d_in[17];
    const float* W_out    = (const float*)d_in[18];
    const float* b_out    = (const float*)d_in[19];
    float* out = (float*)d_out;

    float* w = (float*)d_ws;
    float* ie  = w;                         // 1M floats
    float* pe  = w + 1048576;               // 1M
    float* qb  = w + 2097152;               // 1M
    float* kb  = w + 3145728;               // 2M
    float* vb  = w + 5242880;               // 2M
    float* h1  = w + 7340032;               // 1M  (total 32 MB)
    // reuse:
    float* oc      = pe;    // o_concat     (pe dead after qkv)
    float* attnout = qb;    // out_proj out (q dead after attention)
    float* fmid    = ie;    // FFN mid      (ie dead after h1)
    float* ffnout  = kb;    // FFN out      (k dead)
    float* zb      = vb;    // z            (v dead)
    float* partial = qb;    // final partials

    embed_kernel<<<ROWS / 256, 256, 0, stream>>>(x, protos, W_emb, b_emb,
                                                 W_proto, b_proto, ie, pe);
    qkv_kernel<<<dim3(KV * Bb / 16, 24), 32, 0, stream>>>(ie, pe, in_w, in_b,
                                                          qb, kb, vb);
    attn_kernel<<<dim3(S / 16, Bb * Hh), 32, 0, stream>>>(qb, kb, vb, oc);
    gemm_kernel<<<dim3(ROWS / 16, Ee / 16), 32, 0, stream>>>(oc, op_w, op_b, attnout,
                                                             Ee, Ee, 0);
    lnadd_kernel<<<ROWS / 8, 256, 0, stream>>>(ie, attnout, g1, be1, h1);
    gemm_kernel<<<dim3(ROWS / 16, Ii / 16), 32, 0, stream>>>(h1, W1, b1, fmid,
                                                             Ee, Ii, 1);
    gemm_kernel<<<dim3(ROWS / 16, Ee / 16), 32, 0, stream>>>(fmid, W2, b2, ffnout,
                                                             Ii, Ee, 0);
    lnadd_kernel<<<ROWS / 8, 256, 0, stream>>>(h1, ffnout, g2, be2, zb);
    final_partial<<<dim3(64, 8), 32, 0, stream>>>(zb, W_out, partial);
    final_reduce<<<8, 256, 0, stream>>>(partial, b_out, out);
}